// GraphDecoder_12627203850710
// MI455X (gfx1250) — compile-verified
//
#include <hip/hip_runtime.h>
#include <stdint.h>

#define NNODES 16384
#define NEDGES 262144
#define ETOT   (NEDGES + NNODES)
#define NEG_SLOPE 0.2f

typedef float v2f __attribute__((ext_vector_type(2)));
typedef float v4f __attribute__((ext_vector_type(4)));
typedef float v8f __attribute__((ext_vector_type(8)));

// ---------------------------------------------------------------------------
// CSR build: counts -> exclusive scan -> scatter (src ids, self loops added)
// ---------------------------------------------------------------------------
__global__ void init_kernel(int* cnt, int* cursor) {
    int t = blockIdx.x * blockDim.x + threadIdx.x;
    if (t < NNODES) { cnt[t] = 0; cursor[t] = 0; }
}

__global__ void count_kernel(const long long* __restrict__ adj, int* __restrict__ cnt) {
    int t = blockIdx.x * blockDim.x + threadIdx.x;
    if (t < ETOT) {
        int d = (t < NEDGES) ? (int)adj[NEDGES + t] : (t - NEDGES);
        atomicAdd(&cnt[d], 1);
    }
}

__global__ __launch_bounds__(1024)
void scan_kernel(const int* __restrict__ cnt, int* __restrict__ row_ptr) {
    __shared__ int part[1024];
    int t = threadIdx.x;
    int base = t * 16;
    int loc[16];
    int s = 0;
    #pragma unroll
    for (int i = 0; i < 16; ++i) { loc[i] = s; s += cnt[base + i]; }
    part[t] = s;
    __syncthreads();
    for (int off = 1; off < 1024; off <<= 1) {
        int v = part[t];
        int add = (t >= off) ? part[t - off] : 0;
        __syncthreads();
        part[t] = v + add;
        __syncthreads();
    }
    int excl = (t == 0) ? 0 : part[t - 1];
    #pragma unroll
    for (int i = 0; i < 16; ++i) row_ptr[base + i] = excl + loc[i];
    if (t == 1023) row_ptr[NNODES] = part[1023];
}

__global__ void scatter_kernel(const long long* __restrict__ adj,
                               const int* __restrict__ row_ptr,
                               int* __restrict__ cursor, int* __restrict__ csr_src) {
    int t = blockIdx.x * blockDim.x + threadIdx.x;
    if (t < ETOT) {
        int sidx, d;
        if (t < NEDGES) { sidx = (int)adj[t]; d = (int)adj[NEDGES + t]; }
        else            { sidx = t - NEDGES; d = sidx; }
        int pos = atomicAdd(&cursor[d], 1);
        csr_src[row_ptr[d] + pos] = sidx;
    }
}

// ---------------------------------------------------------------------------
// fp32 WMMA GEMM: C[M,NC] = A[M,K] @ B[K,NC]  (+bias, +relu optional)
// Wave tile 32x64 (2x4 accumulators), WG tile 128x128 (8 waves).
// A/B panels are L2-resident (<=34MB), loads go straight from global.
// ---------------------------------------------------------------------------
template<int K, int NC, bool BIAS, bool RELU>
__global__ __launch_bounds__(256)
void wmma_gemm_kernel(const float* __restrict__ A, const float* __restrict__ B,
                      const float* __restrict__ bias, float* __restrict__ C) {
    const int lane = threadIdx.x & 31;
    const int wave = threadIdx.x >> 5;
    const int wm = wave & 3;           // 0..3 -> M
    const int wn = wave >> 2;          // 0..1 -> N
    const int m0 = blockIdx.x * 128 + wm * 32;
    const int n0 = blockIdx.y * 128 + wn * 64;
    const int lrow = lane & 15;
    const int lhi  = lane >> 4;        // 0/1 -> K pair select

    v8f acc[2][4] = {};

    const float* Arow0 = A + (size_t)(m0 + lrow) * K;
    const float* Arow1 = A + (size_t)(m0 + 16 + lrow) * K;

    for (int k = 0; k < K; k += 4) {
        const int ka = k + 2 * lhi;
        v2f a0 = *(const v2f*)(Arow0 + ka);
        v2f a1 = *(const v2f*)(Arow1 + ka);
        #pragma unroll
        for (int j = 0; j < 4; ++j) {
            const int col = n0 + j * 16 + lrow;
            v2f b;
            b.x = B[(size_t)ka * NC + col];
            b.y = B[(size_t)(ka + 1) * NC + col];
            acc[0][j] = __builtin_amdgcn_wmma_f32_16x16x4_f32(
                false, a0, false, b, (short)0, acc[0][j], false, false);
            acc[1][j] = __builtin_amdgcn_wmma_f32_16x16x4_f32(
                false, a1, false, b, (short)0, acc[1][j], false, false);
        }
    }

    #pragma unroll
    for (int i = 0; i < 2; ++i) {
        #pragma unroll
        for (int j = 0; j < 4; ++j) {
            const int n = n0 + j * 16 + lrow;
            const float bv = BIAS ? bias[n] : 0.0f;
            #pragma unroll
            for (int r = 0; r < 8; ++r) {
                const int m = m0 + i * 16 + lhi * 8 + r;
                float v = acc[i][j][r] + bv;
                if (RELU) v = v > 0.f ? v : 0.f;
                C[(size_t)m * NC + n] = v;
            }
        }
    }
}

// ---------------------------------------------------------------------------
// al_s/al_d: one wave per (node, head) row dot with a_src/a_dst
// ---------------------------------------------------------------------------
template<int HEADS, int CH>
__global__ __launch_bounds__(256)
void rowdot_kernel(const float* __restrict__ H, const float* __restrict__ a_src,
                   const float* __restrict__ a_dst, float* __restrict__ al_s,
                   float* __restrict__ al_d) {
    const int wave = threadIdx.x >> 5;
    const int lane = threadIdx.x & 31;
    const int r = blockIdx.x * 8 + wave;   // < NNODES*HEADS (exact cover)
    const int n = r / HEADS;
    const int h = r % HEADS;
    const float* row = H + (size_t)n * (HEADS * CH) + h * CH;
    const float* as = a_src + h * CH;
    const float* ad = a_dst + h * CH;
    float ps = 0.f, pd = 0.f;
    for (int i = lane; i < CH; i += 32) {
        float v = row[i];
        ps += v * as[i];
        pd += v * ad[i];
    }
    #pragma unroll
    for (int off = 16; off; off >>= 1) {
        ps += __shfl_xor(ps, off, 32);
        pd += __shfl_xor(pd, off, 32);
    }
    if (lane == 0) { al_s[r] = ps; al_d[r] = pd; }
}

// ---------------------------------------------------------------------------
// GAT aggregation: one WG (128 threads) per dst node, no float atomics.
// Phase 1: block-reduce softmax max & denom over incoming edges.
// Phase 2: chunk edges through LDS (src id + alpha), each thread owns a
//          float4 of channels; gathers are coalesced b128 loads from L2.
// Epilogue: out = relu(agg + bias).
// ---------------------------------------------------------------------------
template<int HEADS>
__global__ __launch_bounds__(128)
void agg_kernel(const float* __restrict__ H, const float* __restrict__ al_s,
                const float* __restrict__ al_d, const int* __restrict__ row_ptr,
                const int* __restrict__ csr_src, const float* __restrict__ bias,
                float* __restrict__ out) {
    const int CTOT = 512;
    const int CHD  = CTOT / HEADS;
    __shared__ float red[128];
    __shared__ int   ssrc[128];
    __shared__ float sw[HEADS * 128];
    __shared__ float msh[HEADS], ssh[HEADS];

    const int n = blockIdx.x;
    const int tid = threadIdx.x;
    const int start = row_ptr[n];
    const int deg = row_ptr[n + 1] - start;

    float ald[HEADS];
    #pragma unroll
    for (int h = 0; h < HEADS; ++h) ald[h] = al_d[n * HEADS + h];

    // softmax statistics per head
    for (int h = 0; h < HEADS; ++h) {
        float lm = -3.4e38f;
        for (int i = tid; i < deg; i += 128) {
            int s = csr_src[start + i];
            float e = al_s[s * HEADS + h] + ald[h];
            e = e > 0.f ? e : NEG_SLOPE * e;
            lm = fmaxf(lm, e);
        }
        red[tid] = lm; __syncthreads();
        for (int off = 64; off; off >>= 1) {
            if (tid < off) red[tid] = fmaxf(red[tid], red[tid + off]);
            __syncthreads();
        }
        const float m = red[0];
        __syncthreads();
        float lsum = 0.f;
        for (int i = tid; i < deg; i += 128) {
            int s = csr_src[start + i];
            float e = al_s[s * HEADS + h] + ald[h];
            e = e > 0.f ? e : NEG_SLOPE * e;
            lsum += __expf(e - m);
        }
        red[tid] = lsum; __syncthreads();
        for (int off = 64; off; off >>= 1) {
            if (tid < off) red[tid] += red[tid + off];
            __syncthreads();
        }
        if (tid == 0) { msh[h] = m; ssh[h] = red[0]; }
        __syncthreads();
    }

    const int c0 = 4 * tid;
    const int hidx = c0 / CHD;       // head owning this channel group
    v4f acc = {0.f, 0.f, 0.f, 0.f};

    for (int base = 0; base < deg; base += 128) {
        const int cn = min(128, deg - base);
        if (tid < cn) {
            int s = csr_src[start + base + tid];
            ssrc[tid] = s;
            #pragma unroll
            for (int h = 0; h < HEADS; ++h) {
                float e = al_s[s * HEADS + h] + ald[h];
                e = e > 0.f ? e : NEG_SLOPE * e;
                sw[h * 128 + tid] = __expf(e - msh[h]) / (ssh[h] + 1e-16f);
            }
        }
        __syncthreads();
        for (int j = 0; j < cn; ++j) {
            const int s = ssrc[j];
            const v4f hv = *(const v4f*)(H + (size_t)s * CTOT + c0);
            const float w = sw[hidx * 128 + j];
            acc.x += w * hv.x;
            acc.y += w * hv.y;
            acc.z += w * hv.z;
            acc.w += w * hv.w;
        }
        __syncthreads();
    }

    const v4f bv = *(const v4f*)(bias + c0);
    v4f o;
    o.x = acc.x + bv.x; o.y = acc.y + bv.y; o.z = acc.z + bv.z; o.w = acc.w + bv.w;
    o.x = o.x > 0.f ? o.x : 0.f;
    o.y = o.y > 0.f ? o.y : 0.f;
    o.z = o.z > 0.f ? o.z : 0.f;
    o.w = o.w > 0.f ? o.w : 0.f;
    *(v4f*)(out + (size_t)n * CTOT + c0) = o;
}

// ---------------------------------------------------------------------------
// launch
// ---------------------------------------------------------------------------
extern "C" void kernel_launch(void* const* d_in, const int* in_sizes, int n_in,
                              void* d_out, int out_size, void* d_ws, size_t ws_size,
                              hipStream_t stream) {
    (void)in_sizes; (void)n_in; (void)out_size; (void)ws_size;

    const float*     x      = (const float*)d_in[0];
    const long long* adj    = (const long long*)d_in[1];
    const float*     W_fc   = (const float*)d_in[2];
    const float*     b_fc   = (const float*)d_in[3];
    const float*     W1     = (const float*)d_in[4];
    const float*     a1_src = (const float*)d_in[5];
    const float*     a1_dst = (const float*)d_in[6];
    const float*     b1     = (const float*)d_in[7];
    const float*     W2     = (const float*)d_in[8];
    const float*     a2_src = (const float*)d_in[9];
    const float*     a2_dst = (const float*)d_in[10];
    const float*     b2     = (const float*)d_in[11];
    float*           y      = (float*)d_out;

    char* p = (char*)d_ws;
    auto carve = [&](size_t bytes) -> char* {
        char* q = p;
        p += (bytes + 255) & ~(size_t)255;
        return q;
    };
    int*   cnt     = (int*)carve((size_t)NNODES * 4);
    int*   cursor  = (int*)carve((size_t)NNODES * 4);
    int*   row_ptr = (int*)carve((size_t)(NNODES + 1) * 4);
    int*   csr_src = (int*)carve((size_t)ETOT * 4);
    float* h       = (float*)carve((size_t)NNODES * 256 * 4);
    float* h1      = (float*)carve((size_t)NNODES * 512 * 4);
    float* hb      = (float*)carve((size_t)NNODES * 512 * 4);
    float* h2      = h1;  // h1 dead after agg1; reuse for gat2 features
    float* al1s    = (float*)carve((size_t)NNODES * 2 * 4);
    float* al1d    = (float*)carve((size_t)NNODES * 2 * 4);
    float* al2s    = (float*)carve((size_t)NNODES * 4);
    float* al2d    = (float*)carve((size_t)NNODES * 4);

    // CSR by destination (self loops included)
    init_kernel<<<(NNODES + 255) / 256, 256, 0, stream>>>(cnt, cursor);
    count_kernel<<<(ETOT + 255) / 256, 256, 0, stream>>>(adj, cnt);
    scan_kernel<<<1, 1024, 0, stream>>>(cnt, row_ptr);
    scatter_kernel<<<(ETOT + 255) / 256, 256, 0, stream>>>(adj, row_ptr, cursor, csr_src);

    // fc + ReLU
    wmma_gemm_kernel<256, 256, true, true><<<dim3(128, 2), 256, 0, stream>>>(x, W_fc, b_fc, h);
    // gat1 linear (bias applied after aggregation, per PyG)
    wmma_gemm_kernel<256, 512, false, false><<<dim3(128, 4), 256, 0, stream>>>(h, W1, nullptr, h1);
    rowdot_kernel<2, 256><<<NNODES * 2 / 8, 256, 0, stream>>>(h1, a1_src, a1_dst, al1s, al1d);
    agg_kernel<2><<<NNODES, 128, 0, stream>>>(h1, al1s, al1d, row_ptr, csr_src, b1, hb);
    // gat2 linear
    wmma_gemm_kernel<512, 512, false, false><<<dim3(128, 4), 256, 0, stream>>>(hb, W2, nullptr, h2);
    rowdot_kernel<1, 512><<<NNODES / 8, 256, 0, stream>>>(h2, a2_src, a2_dst, al2s, al2d);
    agg_kernel<1><<<NNODES, 128, 0, stream>>>(h2, al2s, al2d, row_ptr, csr_src, b2, y);
}